// SpecEMA_52793738002704
// MI455X (gfx1250) — compile-verified
//
#include <hip/hip_runtime.h>
#include <math.h>

// Problem dims (fixed by the reference)
#define B_DIM 64
#define C_DIM 2
#define T_DIM 4000
#define F_DIM 96
#define ALPHA_F 0.99f
// T split into chunks for parallelism; chunk must be a multiple of 16.
#define NCHUNK 25
#define CHUNK_LEN 160              // T_DIM / NCHUNK
#define TILES_PER_CHUNK 10         // CHUNK_LEN / 16
#define OUT_ELEMS (B_DIM * C_DIM * T_DIM * F_DIM)
#define CH_STRIDE (T_DIM * F_DIM)  // channel stride in elements (1,536,000 B: fits IOFFSET)

typedef float v2f __attribute__((ext_vector_type(2)));
typedef float v4f __attribute__((ext_vector_type(4)));
typedef float v8f __attribute__((ext_vector_type(8)));

// LDS tile layout: [channel][f][t] with padded row stride (words)
#define LDS_ST 20                  // padded words per f-row (keeps b64/b128 aligned, spreads banks)
#define LDS_CH (16 * LDS_ST)       // words per channel slab
#define LDS_BUF (2 * LDS_CH)       // words per tile buffer (both channels)

// ---------------------------------------------------------------------------
// Phase 1: per-chunk scan contribution with s0 = 0  ->  c_k[b][f]
// (exact by linearity: s_end(chunk) = alpha^L * s_start + c_k)
// Coalesced: 96 threads read contiguous 384B rows per (b,c,t).
// ---------------------------------------------------------------------------
__global__ void spec_ema_phase1(const float* __restrict__ x,
                                float* __restrict__ cchunk) {
  const int f = threadIdx.x;              // 0..95
  const int k = blockIdx.x;               // chunk
  const int b = blockIdx.y;
  const float OMA = (float)(1.0 - 0.99);  // matches reference's (1.0 - ALPHA) cast to f32
  const float* xp = x + (size_t)(b * C_DIM) * T_DIM * F_DIM +
                    (size_t)(k * CHUNK_LEN) * F_DIM + f;
  float s = 0.0f;
#pragma unroll 4
  for (int t = 0; t < CHUNK_LEN; ++t) {
    float a0 = xp[0];                     // channel 0
    float a1 = xp[CH_STRIDE];             // channel 1 (constant immediate offset)
    float a = fmaf(a0, a0, a1 * a1);
    s = fmaf(a, OMA, s * ALPHA_F);        // new_s = abs2*(1-a) + s*a
    xp += F_DIM;
  }
  cchunk[(k * B_DIM + b) * F_DIM + f] = s;
}

// ---------------------------------------------------------------------------
// Phase 2: sequential combine over the 25 chunks (6144 independent chains).
// Emits per-chunk start states and the final_state output.
// ---------------------------------------------------------------------------
__global__ void spec_ema_phase2(const float* __restrict__ state,
                                const float* __restrict__ cchunk,
                                float* __restrict__ sstart,
                                float* __restrict__ out_final,
                                float alphaL) {
  const int id = blockIdx.x * blockDim.x + threadIdx.x;
  if (id >= B_DIM * F_DIM) return;
  const int b = id / F_DIM;
  const int f = id % F_DIM;
  float s = state[f];                     // state is [1,1,F] broadcast over b
#pragma unroll 1
  for (int k = 0; k < NCHUNK; ++k) {
    const int idx = (k * B_DIM + b) * F_DIM + f;
    sstart[idx] = s;
    s = fmaf(alphaL, s, cchunk[idx]);     // S_{k+1} = a^L * S_k + c_k
  }
  out_final[b * F_DIM + f] = s;           // final EMA state
}

// ---------------------------------------------------------------------------
// Phase 3: WMMA blocked scan.  One wave32 per (chunk, f-group of 16, b).
// Per 16-timestep tile:  S = M x A + p (x) carry  via 4x v_wmma_f32_16x16x4_f32
//   M[i][j] = (1-a)*a^(i-j) lower triangular (constant A-operand),
//   A[j][f] = x0[j][f]^2 + x1[j][f]^2        (B-operand, built from LDS),
//   C[i][f] = a^(i+1) * carry[f].
// Tiles stream global->LDS with transposing GLOBAL_LOAD_ASYNC_TO_LDS_B32,
// double-buffered on ASYNCcnt.  All global addressing uses per-lane base
// pointers advanced by 16*F per tile so row/channel deltas fold into the
// 24-bit instruction offsets.
// ---------------------------------------------------------------------------
__global__ void __launch_bounds__(32)
spec_ema_phase3(const float* __restrict__ x,
                const float* __restrict__ sstart,
                float* __restrict__ out) {
  __shared__ __align__(16) float sh[2 * LDS_BUF];

  const int lane = threadIdx.x;           // 0..31
  const int half = lane >> 4;             // 0: lanes 0-15, 1: lanes 16-31
  const int fl = lane & 15;
  const int k  = blockIdx.x;              // chunk
  const int fg = blockIdx.y;              // f-group (0..5)
  const int b  = blockIdx.z;
  const int f0 = fg * 16;
  const float OMA = (float)(1.0 - 0.99);

  // alpha powers a^0..a^16
  float powA[17];
  powA[0] = 1.0f;
#pragma unroll
  for (int i = 1; i <= 16; ++i) powA[i] = powA[i - 1] * ALPHA_F;

  // Constant A operand (16x4 f32 per WMMA): lane row m = fl,
  // column = 4*kb + 2*half + v  (per ISA 32-bit A-matrix layout).
  v2f aop[4];
#pragma unroll
  for (int kb = 0; kb < 4; ++kb) {
#pragma unroll
    for (int v = 0; v < 2; ++v) {
      const int col = 4 * kb + 2 * half + v;
      aop[kb][v] = (col <= fl) ? OMA * powA[fl - col] : 0.0f;
    }
  }

  // carried state entering this chunk (same 16 values in both lane halves)
  float sp = sstart[(k * B_DIM + b) * F_DIM + f0 + fl];

  const int t_base = k * CHUNK_LEN;

  // Per-lane global base pointers (advanced by 16*F per tile):
  //  gsrc: async-copy source, row tau = half of the current tile, channel 0
  //  ob  : output base, row i = 8*half of the current tile, channel 0
  const float* gsrc = x + ((size_t)(b * 2) * T_DIM + t_base + half) * F_DIM + f0 + fl;
  float*       ob   = out + ((size_t)(b * 2) * T_DIM + t_base + 8 * half) * F_DIM + f0 + fl;

  // LDS row bases for the two tile buffers (channel 0; channel 1 handled by
  // the 2ADDR/immediate channel offset of LDS_CH words).
  const float* prow0 = &sh[0 * LDS_BUF + fl * LDS_ST];
  const float* prow1 = &sh[1 * LDS_BUF + fl * LDS_ST];

  // Transposing async copy of one 16t x 16f x 2c tile into LDS buffer `buf`
  // from per-lane source base g.  Each instruction: 32 lanes x 4B; lanes
  // 0-15 cover global row tau=2q (64B contiguous), lanes 16-31 row tau=2q+1.
  // LDS destination layout is [c][f][t] (the transpose happens in flight).
  auto issue = [&](const float* g, int buf) {
#pragma unroll
    for (int c = 0; c < 2; ++c) {
#pragma unroll
      for (int q = 0; q < 8; ++q) {
        const unsigned lds_addr = (unsigned)(size_t)(
            &sh[buf * LDS_BUF + c * LDS_CH + fl * LDS_ST + 2 * q + half]);
        const unsigned long long ga = (unsigned long long)(size_t)(
            g + (size_t)c * CH_STRIDE + (size_t)(2 * q) * F_DIM);
        asm volatile("global_load_async_to_lds_b32 %0, %1, off"
                     :: "v"(lds_addr), "v"(ga) : "memory");
      }
    }
  };

  issue(gsrc, 0);
  const float* gnext = gsrc + 16 * F_DIM;

#pragma unroll 1
  for (int tt = 0; tt < TILES_PER_CHUNK; ++tt) {
    const int buf = tt & 1;
    if (tt + 1 < TILES_PER_CHUNK) {
      issue(gnext, buf ^ 1);                             // prefetch next tile
      gnext += 16 * F_DIM;
      asm volatile("s_wait_asynccnt 16" ::: "memory");   // current tile landed
    } else {
      asm volatile("s_wait_asynccnt 0" ::: "memory");
    }

    const float* p0 = buf ? prow1 : prow0;               // channel 0, this f
    const float* p1 = p0 + LDS_CH;                       // channel 1, this f

    // B operand (4x16 f32 per WMMA): lane holds a[4kb+2*half+v][fl]
    v2f bop[4];
#pragma unroll
    for (int kb = 0; kb < 4; ++kb) {
      const int j0 = 4 * kb + 2 * half;
      v2f q0 = *(const v2f*)(p0 + j0);
      v2f q1 = *(const v2f*)(p1 + j0);
      bop[kb][0] = fmaf(q0[0], q0[0], q1[0] * q1[0]);
      bop[kb][1] = fmaf(q0[1], q0[1], q1[1] * q1[1]);
    }

    // C = a^(i+1) * carry   (i = r + 8*half per C/D layout)
    v8f acc = {0, 0, 0, 0, 0, 0, 0, 0};
#pragma unroll
    for (int r = 0; r < 8; ++r) acc[r] = powA[r + 8 * half + 1] * sp;

    // D = M x A_tile + C  : four chained f32 WMMAs (K=4 each)
#pragma unroll
    for (int kb = 0; kb < 4; ++kb)
      acc = __builtin_amdgcn_wmma_f32_16x16x4_f32(
          false, aop[kb], false, bop[kb], (short)0, acc, false, false);

    // carry out: row i=15 lives in acc[7] of lanes 16..31 -> broadcast
    sp = __shfl(acc[7], 16 + fl, 32);

    // normalize: y = x * rsq(s)  (v_rsq_f32, ~1-2 ulp; reference is x/sqrt(s))
    v4f xa0 = *(const v4f*)(p0 + 8 * half);
    v4f xb0 = *(const v4f*)(p0 + 8 * half + 4);
    v4f xa1 = *(const v4f*)(p1 + 8 * half);
    v4f xb1 = *(const v4f*)(p1 + 8 * half + 4);
#pragma unroll
    for (int r = 0; r < 8; ++r) {
      const float inv = __builtin_amdgcn_rsqf(acc[r]);   // v_rsq_f32
      const float x0v = (r < 4) ? xa0[r] : xb0[r - 4];
      const float x1v = (r < 4) ? xa1[r] : xb1[r - 4];
      ob[r * F_DIM]             = x0v * inv;             // const IOFFSET: r*384B
      ob[r * F_DIM + CH_STRIDE] = x1v * inv;             // + 1,536,000B (fits 24b)
    }
    ob += 16 * F_DIM;                                    // advance one tile
  }
}

// ---------------------------------------------------------------------------
extern "C" void kernel_launch(void* const* d_in, const int* in_sizes, int n_in,
                              void* d_out, int out_size, void* d_ws, size_t ws_size,
                              hipStream_t stream) {
  (void)in_sizes; (void)n_in; (void)out_size; (void)ws_size;
  const float* x     = (const float*)d_in[0];   // [B,C,T,F] f32
  const float* state = (const float*)d_in[1];   // [1,1,F]   f32
  float* out = (float*)d_out;                   // [B,C,T,F] ++ [B,1,F]

  float* cchunk = (float*)d_ws;                            // NCHUNK*B*F f32
  float* sstart = cchunk + (size_t)NCHUNK * B_DIM * F_DIM; // NCHUNK*B*F f32
  const float alphaL = (float)pow(0.99, (double)CHUNK_LEN);

  spec_ema_phase1<<<dim3(NCHUNK, B_DIM), 96, 0, stream>>>(x, cchunk);
  spec_ema_phase2<<<(B_DIM * F_DIM + 255) / 256, 256, 0, stream>>>(
      state, cchunk, sstart, out + (size_t)OUT_ELEMS, alphaL);
  spec_ema_phase3<<<dim3(NCHUNK, F_DIM / 16, B_DIM), 32, 0, stream>>>(
      x, sstart, out);
}